// FeatureRestrain_43361989820656
// MI455X (gfx1250) — compile-verified
//
#include <hip/hip_runtime.h>

// CDNA5 / gfx1250. wave32.
typedef __attribute__((ext_vector_type(2))) float v2f;
typedef __attribute__((ext_vector_type(8))) float v8f;

#define HW     196      // 14*14 spatial
#define NCH    2048
#define NB     64
#define RANKLE 411u     // 2048 - 1638 + 1 : rank (ascending) of min(top-k)

// ---------------------------------------------------------------------------
// Kernel 1: global average pool via V_WMMA_F32_16X16X4_F32.
// One wave reduces 16 contiguous channel rows (16 x 196 f32 = 12544 B).
// A-matrix (16x4 f32): lane L holds row M = L&15; lanes 0-15 carry K={0,1},
// lanes 16-31 carry K={2,3} (ISA 7.12.2 layout). B = ones (4x16) -> every
// lane's B regs are 1.0f, so D[m][*] accumulates the row sum of A.
// 49 accumulating WMMAs cover K = 196 exactly.
// ---------------------------------------------------------------------------
__global__ __launch_bounds__(256)
void gap_wmma_kernel(const float* __restrict__ in, float* __restrict__ fv)
{
    const int lane  = threadIdx.x & 31;
    const int wid   = (blockIdx.x * blockDim.x + threadIdx.x) >> 5;   // global wave id
    const long cbase = (long)wid * 16;                                 // first channel row of tile

    const int m     = lane & 15;            // matrix row this lane feeds
    const int khalf = (lane >> 4) << 1;     // 0 for lanes 0-15, 2 for lanes 16-31

    const float* __restrict__ rowp = in + cbase * HW + (long)m * HW + khalf;

    v2f bones; bones.x = 1.0f; bones.y = 1.0f;   // B = ones everywhere
    v8f acc = {};                                 // C/D accumulator (16x16 f32)

    #pragma unroll 7
    for (int k0 = 0; k0 < HW; k0 += 4) {
        v2f a = *(const v2f*)(rowp + k0);         // 8B aligned: even float offset
        // D = A(16x4) * ones(4x16) + C  ->  per-row partial sums, all N equal
        acc = __builtin_amdgcn_wmma_f32_16x16x4_f32(
                  /*neg_a=*/false, a, /*neg_b=*/false, bones,
                  /*c_mod=*/(short)0, acc, /*reuse_a=*/false, /*reuse_b=*/false);
    }

    // D layout: VGPR j, lanes 0-15 -> (M=j, N=lane); lanes 16-31 -> (M=j+8).
    // All N columns identical => lane 0 holds sums for channels m=0..7,
    // lane 16 holds sums for channels m=8..15.
    const float inv = 1.0f / (float)HW;
    if (lane == 0 || lane == 16) {
        float* o = fv + cbase + ((lane >> 4) << 3);
        float4 w0, w1;
        w0.x = acc[0] * inv; w0.y = acc[1] * inv;
        w0.z = acc[2] * inv; w0.w = acc[3] * inv;
        w1.x = acc[4] * inv; w1.y = acc[5] * inv;
        w1.z = acc[6] * inv; w1.w = acc[7] * inv;
        *(float4*)(o)     = w0;
        *(float4*)(o + 4) = w1;
    }
}

// ---------------------------------------------------------------------------
// Kernel 2: per-sample exact k-th-largest selection + mask, in place on d_out.
// Block b owns sample b. 256 threads x 8 values, kept in registers as
// order-preserving uint keys. 32-step binary search over key space finds the
// smallest key u with count(keys <= u) >= 411; that key IS the 411th-smallest
// element = min(top-1638). Mask = (key >= kth) ? ALPHA : BETA.
// ---------------------------------------------------------------------------
__device__ __forceinline__ unsigned f2key(float f) {
    unsigned u = __float_as_uint(f);
    return u ^ ((u >> 31) ? 0xFFFFFFFFu : 0x80000000u);   // monotone map
}

__global__ __launch_bounds__(256)
void select_mask_kernel(float* __restrict__ out)
{
    const int b = blockIdx.x;
    const int t = threadIdx.x;
    float* row = out + (long)b * NCH;

    float4 f0 = *(const float4*)(row + t * 8);
    float4 f1 = *(const float4*)(row + t * 8 + 4);

    unsigned k[8];
    k[0] = f2key(f0.x); k[1] = f2key(f0.y); k[2] = f2key(f0.z); k[3] = f2key(f0.w);
    k[4] = f2key(f1.x); k[5] = f2key(f1.y); k[6] = f2key(f1.z); k[7] = f2key(f1.w);

    __shared__ unsigned cnt;
    unsigned lo = 0u, hi = 0xFFFFFFFFu;
    while (lo < hi) {                       // uniform across block
        unsigned mid = lo + ((hi - lo) >> 1);
        if (t == 0) cnt = 0u;
        __syncthreads();
        unsigned c = 0;
        #pragma unroll
        for (int i = 0; i < 8; ++i) c += (k[i] <= mid) ? 1u : 0u;
        atomicAdd(&cnt, c);
        __syncthreads();
        unsigned total = cnt;
        __syncthreads();                    // all threads read cnt before reuse
        if (total >= RANKLE) hi = mid; else lo = mid + 1u;
    }
    const unsigned kth = lo;                // key of min(top-k), exact

    float4 o0, o1;
    o0.x = (k[0] >= kth) ? 0.8f : 1.2f;
    o0.y = (k[1] >= kth) ? 0.8f : 1.2f;
    o0.z = (k[2] >= kth) ? 0.8f : 1.2f;
    o0.w = (k[3] >= kth) ? 0.8f : 1.2f;
    o1.x = (k[4] >= kth) ? 0.8f : 1.2f;
    o1.y = (k[5] >= kth) ? 0.8f : 1.2f;
    o1.z = (k[6] >= kth) ? 0.8f : 1.2f;
    o1.w = (k[7] >= kth) ? 0.8f : 1.2f;
    // each thread overwrites exactly the slots it loaded -> no cross-thread hazard
    *(float4*)(row + t * 8)     = o0;
    *(float4*)(row + t * 8 + 4) = o1;
}

// ---------------------------------------------------------------------------
extern "C" void kernel_launch(void* const* d_in, const int* in_sizes, int n_in,
                              void* d_out, int out_size, void* d_ws, size_t ws_size,
                              hipStream_t stream)
{
    (void)in_sizes; (void)n_in; (void)out_size; (void)d_ws; (void)ws_size;
    const float* in = (const float*)d_in[0];
    float* out = (float*)d_out;

    // 64*2048 = 131072 channel rows -> 8192 waves of 16 rows -> 1024 blocks x 8 waves
    gap_wmma_kernel<<<dim3(1024), dim3(256), 0, stream>>>(in, out);
    // one block per sample
    select_mask_kernel<<<dim3(NB), dim3(256), 0, stream>>>(out);
}